// Integrate_feats_module_43387759624384
// MI455X (gfx1250) — compile-verified
//
#include <hip/hip_runtime.h>
#include <hip/hip_bf16.h>

typedef __attribute__((ext_vector_type(2))) float v2f;
typedef __attribute__((ext_vector_type(8))) float v8f;

#define BS    256
#define C     2048
#define HW    128     // h*w = 16*8
#define D     128
#define CHUNK 64

// ---------------------------------------------------------------------------
// Kernel 1: partial[part][b][hw] = sum_{c in part} clsW[pids[b]][c]*feats[b][c][hw]
// HBM-bound: streams 268 MB. nparts blocks per b for deep MLP; per wave-iter:
// 4x global_load_b128 (immediate offsets) + 1x ds_load_b128 + 16 FMA + 1 prefetch
// 64 KB ahead. 256 threads = 8 waves; wave wv owns 4-row groups strided by 32.
// ---------------------------------------------------------------------------
__global__ __launch_bounds__(256) void k_weighted(const float* __restrict__ feats,
                                                  const long long* __restrict__ pids,
                                                  const float* __restrict__ clsW,
                                                  float* __restrict__ partial,
                                                  int nparts) {
    __shared__ float s_w[C];        // classifier slice (worst case 8 KB)
    __shared__ float s_red[8 * HW]; // 4 KB per-wave partials

    const int b    = blockIdx.x / nparts;
    const int part = blockIdx.x % nparts;
    const int rows = C / nparts;
    const int tid  = threadIdx.x;

    const int pid = (int)pids[b];
    const float* wrow = clsW + (size_t)pid * C + (size_t)part * rows;
    for (int i = tid; i < rows; i += 256) s_w[i] = wrow[i];
    __syncthreads();

    const float* fb = feats + ((size_t)b * C + (size_t)part * rows) * HW;
    const int lane = tid & 31;
    const int wv   = tid >> 5;
    const int hw4  = lane * 4;

    float4 acc = make_float4(0.f, 0.f, 0.f, 0.f);
    for (int base = wv * 4; base < rows; base += 32) {
        if (base + 131 < rows)  // 64 KB ahead; lane*64B strides cover all 4 rows
            __builtin_prefetch(fb + (size_t)(base + 128) * HW + lane * 16, 0, 1);
        const float* p = fb + (size_t)base * HW + hw4;
        const float4 f0 = *(const float4*)(p);
        const float4 f1 = *(const float4*)(p + HW);
        const float4 f2 = *(const float4*)(p + 2 * HW);
        const float4 f3 = *(const float4*)(p + 3 * HW);
        const float4 w4 = *(const float4*)&s_w[base];   // wave-uniform ds_load_b128
        acc.x += w4.x * f0.x + w4.y * f1.x + w4.z * f2.x + w4.w * f3.x;
        acc.y += w4.x * f0.y + w4.y * f1.y + w4.z * f2.y + w4.w * f3.y;
        acc.z += w4.x * f0.z + w4.y * f1.z + w4.z * f2.z + w4.w * f3.z;
        acc.w += w4.x * f0.w + w4.y * f1.w + w4.z * f2.w + w4.w * f3.w;
    }
    s_red[wv * HW + hw4 + 0] = acc.x;
    s_red[wv * HW + hw4 + 1] = acc.y;
    s_red[wv * HW + hw4 + 2] = acc.z;
    s_red[wv * HW + hw4 + 3] = acc.w;
    __syncthreads();

    if (tid < HW) {
        float s = 0.f;
        #pragma unroll
        for (int w2 = 0; w2 < 8; ++w2) s += s_red[w2 * HW + tid];
        partial[((size_t)part * BS + b) * HW + tid] = s;
    }
}

// ---------------------------------------------------------------------------
// Kernel 2: Wh = x @ W with x = sum of c-partials folded into the A-load.
// NPARTS is a template constant -> branch-free clause loads at fixed 128 KB
// strides. One wave per 16x16 tile; 32x v_wmma_f32_16x16x4_f32 over K=128.
// A 16x4 f32 layout: lanes 0-15 -> K={kb,kb+1}, lanes 16-31 -> K={kb+2,kb+3}.
// C/D 16x16 f32 layout: VGPR v, lanes 0-15 -> M=v, lanes 16-31 -> M=v+8.
// ---------------------------------------------------------------------------
template <int NPARTS>
__global__ __launch_bounds__(128) void k_gemm_wmma(const float* __restrict__ parts,
                                                   const float* __restrict__ W,
                                                   float* __restrict__ Wh) {
    const int wave = blockIdx.x * 4 + (threadIdx.x >> 5); // 0..127
    const int lane = threadIdx.x & 31;
    const int tM = wave >> 3;                 // 16 M tiles over 256 rows
    const int tN = wave & 7;                  // 8 N tiles over 128 cols
    const int hi = lane >> 4;
    const int row = tM * 16 + (lane & 15);
    const int col = tN * 16 + (lane & 15);

    v8f acc = {};
    #pragma unroll 4
    for (int kb = 0; kb < D; kb += 4) {
        const int k0 = kb + 2 * hi;
        float ax = 0.f, ay = 0.f;
        #pragma unroll
        for (int p = 0; p < NPARTS; ++p) {
            const float2 t = *(const float2*)(parts + ((size_t)p * BS + row) * D + k0);
            ax += t.x; ay += t.y;
        }
        v2f av; av.x = ax; av.y = ay;
        v2f bv;
        bv.x = W[k0 * D + col];
        bv.y = W[(k0 + 1) * D + col];
        acc = __builtin_amdgcn_wmma_f32_16x16x4_f32(
            /*neg_a=*/false, av, /*neg_b=*/false, bv,
            /*c_mod=*/(short)0, acc, /*reuse_a=*/false, /*reuse_b=*/false);
    }
    #pragma unroll
    for (int v = 0; v < 8; ++v) {
        const int mr = tM * 16 + v + 8 * hi;
        Wh[mr * D + col] = acc[v];
    }
}

// ---------------------------------------------------------------------------
// Kernel 3: attention epilogue. 64 blocks x 128 threads (one per y column).
// ---------------------------------------------------------------------------
__global__ __launch_bounds__(128) void k_attn(const float* __restrict__ Wh,
                                              const float* __restrict__ a,
                                              const long long* __restrict__ pids,
                                              float* __restrict__ out) {
    __shared__ float red[8][HW];
    const int b = blockIdx.x;
    const int y = threadIdx.x;

    float wh[4];
    #pragma unroll
    for (int n = 0; n < 4; ++n) wh[n] = Wh[(b * 4 + n) * D + y];
    const float ai = a[y];
    const float aj = a[D + y];
    #pragma unroll
    for (int n = 0; n < 4; ++n) { red[n][y] = wh[n] * ai; red[4 + n][y] = wh[n] * aj; }
    __syncthreads();
    for (int stride = 64; stride > 0; stride >>= 1) {
        if (y < stride) {
            #pragma unroll
            for (int v = 0; v < 8; ++v) red[v][y] += red[v][y + stride];
        }
        __syncthreads();
    }
    float e_i[4], e_j[4];
    #pragma unroll
    for (int n = 0; n < 4; ++n) { e_i[n] = red[n][0]; e_j[n] = red[4 + n][0]; }

    float o[4];
    #pragma unroll
    for (int i = 0; i < 4; ++i) {
        float l[4], mx = -1e30f;
        #pragma unroll
        for (int j = 0; j < 4; ++j) {
            const float e = e_i[i] + e_j[j];
            l[j] = (e >= 0.f) ? e : 0.01f * e;       // leaky_relu(0.01)
            mx = fmaxf(mx, l[j]);
        }
        float s = 0.f, w4[4];
        #pragma unroll
        for (int j = 0; j < 4; ++j) { w4[j] = expf(l[j] - mx); s += w4[j]; }
        const float inv = 1.f / s;
        float oc = 0.f;
        #pragma unroll
        for (int j = 0; j < 4; ++j) oc += (w4[j] * inv) * wh[j]; // att @ Wh
        o[i] = (oc > 0.f) ? oc : (expf(oc) - 1.f);              // elu
    }
    const float mx = fmaxf(fmaxf(o[0], o[1]), fmaxf(o[2], o[3]));
    float s = 0.f;
    #pragma unroll
    for (int i = 0; i < 4; ++i) s += expf(o[i] - mx);
    const float lse = logf(s) + mx;
    float mean = 0.f;
    #pragma unroll
    for (int i = 0; i < 4; ++i) mean += (o[i] - lse);
    out[b * D + y] = mean * 0.25f;

    if (y == 0) {  // integrate_pids = pids[::4] (int64) after 8192 f32 outputs
        long long* op = (long long*)(out + CHUNK * D);
        op[b] = pids[b * 4];
    }
}

// ---------------------------------------------------------------------------
extern "C" void kernel_launch(void* const* d_in, const int* in_sizes, int n_in,
                              void* d_out, int out_size, void* d_ws, size_t ws_size,
                              hipStream_t stream) {
    (void)in_sizes; (void)n_in; (void)out_size;
    const float*     feats = (const float*)d_in[0];
    const long long* pids  = (const long long*)d_in[1];
    const float*     clsW  = (const float*)d_in[2];
    const float*     Wm    = (const float*)d_in[3];
    const float*     av    = (const float*)d_in[4];
    // d_in[5] = num_same_id (== 4, baked into shapes)

    const size_t unit = (size_t)BS * HW * sizeof(float);  // 128 KB
    // nparts c-split partials + Wh must fit in ws
    const int nparts = (ws_size >= 5 * unit) ? 4 : (ws_size >= 3 * unit) ? 2 : 1;

    float* ws      = (float*)d_ws;
    float* partial = ws;                             // nparts * 256*128 f32
    float* Wh      = ws + (size_t)nparts * BS * HW;  // 256*128 f32
    float* outf    = (float*)d_out;

    k_weighted<<<BS * nparts, 256, 0, stream>>>(feats, pids, clsW, partial, nparts);
    if (nparts == 4)      k_gemm_wmma<4><<<32, 128, 0, stream>>>(partial, Wm, Wh);
    else if (nparts == 2) k_gemm_wmma<2><<<32, 128, 0, stream>>>(partial, Wm, Wh);
    else                  k_gemm_wmma<1><<<32, 128, 0, stream>>>(partial, Wm, Wh);
    k_attn<<<CHUNK, 128, 0, stream>>>(Wh, av, pids, outf);
}